// TurkishAwareLoss_58428735095265
// MI455X (gfx1250) — compile-verified
//
#include <hip/hip_runtime.h>
#include <hip/hip_bf16.h>
#include <math.h>

#define VOCAB 512
#define WAVE 32
#define BLOCK_THREADS 256
#define WAVES_PER_BLOCK (BLOCK_THREADS / WAVE)
#define NBLOCKS 2048

// Probe for the CDNA5 async global->LDS copy builtin (ASYNCcnt path,
// CDNA5 ISA). Fall back to direct b128 loads if absent.
#if defined(__has_builtin)
#if __has_builtin(__builtin_amdgcn_global_load_async_to_lds_b128)
#define USE_ASYNC_LDS 1
#endif
#endif
#ifndef USE_ASYNC_LDS
#define USE_ASYNC_LDS 0
#endif

typedef int v4i __attribute__((ext_vector_type(4)));
typedef __attribute__((address_space(1))) v4i GlobalV4i;  // global (AS1) int4
typedef __attribute__((address_space(3))) v4i LdsV4i;     // LDS (AS3) int4

__device__ __forceinline__ void wait_async_le4() {
#if defined(__has_builtin) && __has_builtin(__builtin_amdgcn_s_wait_asynccnt)
  __builtin_amdgcn_s_wait_asynccnt(4);
#else
  asm volatile("s_wait_asynccnt 0x4" ::: "memory");
#endif
}
__device__ __forceinline__ void wait_async_le0() {
#if defined(__has_builtin) && __has_builtin(__builtin_amdgcn_s_wait_asynccnt)
  __builtin_amdgcn_s_wait_asynccnt(0);
#else
  asm volatile("s_wait_asynccnt 0x0" ::: "memory");
#endif
}

// SIMILAR_GROUPS = ['ıi'(1), 'oö'(2), 'uü'(3), 'gğ'(4)]
// codepoints: i=105, ı=305 | o=111, ö=246 | u=117, ü=252 | g=103, ğ=287
__device__ __forceinline__ int grp(int c) {
  if (c == 105 || c == 305) return 1;
  if (c == 111 || c == 246) return 2;
  if (c == 117 || c == 252) return 3;
  if (c == 103 || c == 287) return 4;
  return 0;
}

__global__ __launch_bounds__(BLOCK_THREADS)
void turkish_ce_rows(const float* __restrict__ pred,
                     const int* __restrict__ target,
                     int nrows,
                     float* __restrict__ partials) {
#if USE_ASYNC_LDS
  __shared__ float ldsbuf[WAVES_PER_BLOCK * 2 * VOCAB];  // 8 waves * 2 bufs * 2KB = 32KB
#endif
  __shared__ float wsum[WAVES_PER_BLOCK];

  const int lane = threadIdx.x & (WAVE - 1);
  const int wib  = threadIdx.x >> 5;                       // wave index in block
  const int gw   = blockIdx.x * WAVES_PER_BLOCK + wib;     // global wave id
  const int nw   = gridDim.x * WAVES_PER_BLOCK;            // total waves

  float wacc = 0.0f;

#if USE_ASYNC_LDS
  // Issue 4 async b128 copies: 32 lanes * 16B = 512B each -> one 2KB row.
  // Lane l, chunk c covers elements idx = c*128 + l*4 .. +3 (coalesced).
  auto issue_row = [&](int row, int buf) {
    const float* g = pred + (size_t)row * VOCAB + lane * 4;
    float* l = ldsbuf + (size_t)(wib * 2 + buf) * VOCAB + lane * 4;
#pragma unroll
    for (int c = 0; c < 4; ++c) {
      __builtin_amdgcn_global_load_async_to_lds_b128(
          (GlobalV4i*)(g + c * 128), (LdsV4i*)(l + c * 128), 0, 0);
    }
  };
#endif

  int row = gw;
  int parity = 0;
#if USE_ASYNC_LDS
  if (row < nrows) issue_row(row, 0);
#endif

  for (; row < nrows; row += nw) {
    float4 v[4];
#if USE_ASYNC_LDS
    const int nxt = row + nw;
    if (nxt < nrows) {
      issue_row(nxt, parity ^ 1);
      wait_async_le4();   // current row's 4 copies done (in-order per wave)
    } else {
      wait_async_le0();
    }
    {
      const float* l = ldsbuf + (size_t)(wib * 2 + parity) * VOCAB + lane * 4;
#pragma unroll
      for (int c = 0; c < 4; ++c) v[c] = *(const float4*)(l + c * 128);
    }
    parity ^= 1;
#else
    {
      const float* g = pred + (size_t)row * VOCAB + lane * 4;
#pragma unroll
      for (int c = 0; c < 4; ++c) v[c] = *(const float4*)(g + c * 128);
    }
#endif

    // ---- pass 1: max + argmax (first-index tie-break like jnp.argmax) ----
    float mx = v[0].x;
    int mi = lane * 4;
#pragma unroll
    for (int c = 0; c < 4; ++c) {
      const float vals[4] = {v[c].x, v[c].y, v[c].z, v[c].w};
#pragma unroll
      for (int k = 0; k < 4; ++k) {
        const int idx = c * 128 + lane * 4 + k;
        const float x = vals[k];
        if (x > mx) { mx = x; mi = idx; }   // ascending idx within lane keeps first
      }
    }
#pragma unroll
    for (int off = 16; off > 0; off >>= 1) {
      const float omx = __shfl_xor(mx, off, WAVE);
      const int   omi = __shfl_xor(mi, off, WAVE);
      if (omx > mx || (omx == mx && omi < mi)) { mx = omx; mi = omi; }
    }

    // ---- pass 2: sum(exp(x - mx)) and gather x[target] (from registers) ----
    const int t = target[row];
    float s = 0.0f, tv = 0.0f;
#pragma unroll
    for (int c = 0; c < 4; ++c) {
      const float vals[4] = {v[c].x, v[c].y, v[c].z, v[c].w};
#pragma unroll
      for (int k = 0; k < 4; ++k) {
        const int idx = c * 128 + lane * 4 + k;
        const float x = vals[k];
        s += expf(x - mx);
        if (idx == t) tv = x;
      }
    }
#pragma unroll
    for (int off = 16; off > 0; off >>= 1) {
      s  += __shfl_xor(s,  off, WAVE);
      tv += __shfl_xor(tv, off, WAVE);
    }

    float loss = mx + logf(s) - tv;                 // = -log_softmax(pred)[t]
    const int gp = grp(mi), gt = grp(t);
    if (gp > 0 && gp == gt) loss *= 0.8f;
    wacc += loss;                                   // uniform across lanes
  }

  if (lane == 0) wsum[wib] = wacc;
  __syncthreads();
  if (threadIdx.x == 0) {
    float b = 0.0f;
#pragma unroll
    for (int i = 0; i < WAVES_PER_BLOCK; ++i) b += wsum[i];
    partials[blockIdx.x] = b;                       // deterministic per-block partial
  }
}

__global__ __launch_bounds__(256)
void turkish_ce_finalize(const float* __restrict__ partials, int nparts,
                         int nrows, float* __restrict__ out) {
  __shared__ double sd[256];
  double a = 0.0;
  for (int i = threadIdx.x; i < nparts; i += 256) a += (double)partials[i];
  sd[threadIdx.x] = a;
  __syncthreads();
#pragma unroll
  for (int s = 128; s > 0; s >>= 1) {
    if ((int)threadIdx.x < s) sd[threadIdx.x] += sd[threadIdx.x + s];
    __syncthreads();
  }
  if (threadIdx.x == 0) out[0] = (float)(sd[0] / (double)nrows);
}

extern "C" void kernel_launch(void* const* d_in, const int* in_sizes, int n_in,
                              void* d_out, int out_size, void* d_ws, size_t ws_size,
                              hipStream_t stream) {
  (void)n_in; (void)out_size; (void)ws_size;
  const float* pred   = (const float*)d_in[0];
  const int*   target = (const int*)d_in[1];
  float*       out    = (float*)d_out;
  float*       parts  = (float*)d_ws;

  const int nrows = in_sizes[0] / VOCAB;  // 524288

  turkish_ce_rows<<<NBLOCKS, BLOCK_THREADS, 0, stream>>>(pred, target, nrows, parts);
  turkish_ce_finalize<<<1, 256, 0, stream>>>(parts, NBLOCKS, nrows, out);
}